// StatsQuantizer_specific_4_qkreparam_cga_34772055229136
// MI455X (gfx1250) — compile-verified
//
#include <hip/hip_runtime.h>
#include <math.h>

// CDNA5 / gfx1250: wave32, WMMA, async global->LDS.
typedef float v2f __attribute__((ext_vector_type(2)));
typedef float v8f __attribute__((ext_vector_type(8)));

#define ROW_LEN  8192
#define THREADS  256
#define WAVES    (THREADS / 32)
#define CHUNK    (ROW_LEN / WAVES)   // 1024 floats per wave

__device__ __forceinline__ float quantize1(float w, float inv_scale,
                                           float lo, float hi, float qmul) {
    float s = w * inv_scale;                       // W / sf
    s = fminf(fmaxf(s, lo), hi);                   // clip(-hc, hc-eps)
    float b4 = __builtin_fmaf(s, 8.0f, -0.5f);     // *n - 0.5
    float r  = rintf(b4);                          // round-half-even (v_rndne_f32)
    return (r + 0.5f) * qmul;                      // sf * (r+0.5)/8
}

__global__ __launch_bounds__(THREADS)
void StatsQuantizer_row_kernel(const float* __restrict__ weight,
                               const float* __restrict__ clip_val,
                               float* __restrict__ out) {
    __shared__ float lds_row[ROW_LEN];     // 32 KB: the whole row, HBM-read once
    __shared__ float wave_sums[WAVES];

    const int row  = blockIdx.x;
    const int tid  = threadIdx.x;
    const int wave = tid >> 5;
    const int lane = tid & 31;

    const float* grow = weight + (size_t)row * ROW_LEN;

    // ---------------- Stage: async global -> LDS (ASYNCcnt path) -------------
    // Each wave stages its own 4 KB chunk: 8 x (32 lanes x 16B) = 4096 B.
    const unsigned lds_base   = (unsigned)(unsigned long long)&lds_row[0];
    const unsigned chunk_byte = (unsigned)(wave * CHUNK * 4);
    #pragma unroll
    for (int i = 0; i < (CHUNK * 4) / (32 * 16); ++i) {
        unsigned off = chunk_byte + (unsigned)i * 512u + (unsigned)lane * 16u;
        unsigned dst = lds_base + off;
        asm volatile("global_load_async_to_lds_b128 %0, %1, %2"
                     :: "v"(dst), "v"(off), "s"(grow) : "memory");
    }
    asm volatile("s_wait_asynccnt 0" ::: "memory");
    // Each wave only reads the chunk it loaded itself -> no barrier needed yet.

    // ---------------- Reduce: sum |w| over chunk via WMMA f32 16x16x4 --------
    // Treat the chunk as 16x4 fp32 A-tiles (lane mapping irrelevant for a total
    // sum) and multiply by an all-ones B: D[m][n] accumulates row sums.
    v8f acc = {};
    v2f ones; ones[0] = 1.0f; ones[1] = 1.0f;
    const float* lchunk = &lds_row[wave * CHUNK];
    #pragma unroll 4
    for (int it = 0; it < CHUNK / 64; ++it) {      // 16 iterations, 64 floats each
        float2 d = ((const float2*)lchunk)[it * 32 + lane];  // ds_load_b64
        v2f a;
        a[0] = __builtin_fabsf(d.x);
        a[1] = __builtin_fabsf(d.y);
        acc = __builtin_amdgcn_wmma_f32_16x16x4_f32(
                  false, a, false, ones, (short)0, acc, false, false);
    }
    // Columns of D are identical; total = sum over M of D[M][n] for any n.
    float partial = acc[0] + acc[1] + acc[2] + acc[3] +
                    acc[4] + acc[5] + acc[6] + acc[7];   // M = 0..7 (or 8..15)
    float chunk_sum = partial + __shfl_xor(partial, 16, 32); // add M=8..15 half
    if (lane == 0) wave_sums[wave] = chunk_sum;
    __syncthreads();

    float rowsum = 0.0f;
    #pragma unroll
    for (int i = 0; i < WAVES; ++i) rowsum += wave_sums[i];

    // scaling_factor = 2 * mean(|row|) = rowsum / 4096
    const float scale     = rowsum * (2.0f / (float)ROW_LEN);
    const float inv_scale = 1.0f / scale;          // one divide per row
    const float half_clip = 0.5f * clip_val[0];    // uniform scalar load
    const float hi        = half_clip - 1e-6f;
    const float lo        = -half_clip;
    const float qmul      = scale * 0.125f;

    // ---------------- Quantize from LDS, b128 stores to HBM ------------------
    float* orow = out + (size_t)row * ROW_LEN;
    #pragma unroll
    for (int it = 0; it < CHUNK / (32 * 4); ++it) {   // 8 iterations of float4
        int idx = wave * CHUNK + it * 128 + lane * 4;
        float4 w = *(const float4*)&lds_row[idx];     // ds_load_b128
        float4 q;
        q.x = quantize1(w.x, inv_scale, lo, hi, qmul);
        q.y = quantize1(w.y, inv_scale, lo, hi, qmul);
        q.z = quantize1(w.z, inv_scale, lo, hi, qmul);
        q.w = quantize1(w.w, inv_scale, lo, hi, qmul);
        *(float4*)&orow[idx] = q;                     // global_store_b128
    }
}

extern "C" void kernel_launch(void* const* d_in, const int* in_sizes, int n_in,
                              void* d_out, int out_size, void* d_ws, size_t ws_size,
                              hipStream_t stream) {
    const float* weight = (const float*)d_in[0];   // (8192, 8192) fp32
    const float* clip   = (const float*)d_in[1];   // (1,) fp32
    float* out = (float*)d_out;

    const int rows = in_sizes[0] / ROW_LEN;        // 8192
    hipLaunchKernelGGL(StatsQuantizer_row_kernel,
                       dim3(rows), dim3(THREADS), 0, stream,
                       weight, clip, out);
}